// EdgeDecoder_67259187855552
// MI455X (gfx1250) — compile-verified
//
#include <hip/hip_runtime.h>
#include <math.h>

typedef float v2f __attribute__((ext_vector_type(2)));
typedef float v8f __attribute__((ext_vector_type(8)));

static constexpr int kN = 2048;   // nodes
static constexpr int kD = 256;    // node dim
static constexpr int kH = 64;     // hidden

// ---------------------------------------------------------------------------
// Kernel A: P = X @ [Wi | Wj] via V_WMMA_F32_16X16X4_F32 (one wave = one
// 16x16 tile of the 2048x128 output). b1 folded into the pi half. Results
// stored transposed: piT[h][node] = (X@Wi)[node][h] + b1[h]; pjT[h][node].
//
// WMMA f32 16x16x4 fragment layouts (wave32, ISA 7.12.2):
//   A (16x4): lane L -> row m=L%16, VGPR r holds K = 2*(L>>4)+r  (contig pair)
//   B (4x16): lane L -> col n=L%16, VGPR r holds K = 2*(L>>4)+r
//   C (16x16): lane L -> col n=L%16, VGPR v holds row m = v + 8*(L>>4)
// ---------------------------------------------------------------------------
__global__ __launch_bounds__(32)
void edgedec_gemm_wmma(const float* __restrict__ X,
                       const float* __restrict__ W1,
                       const float* __restrict__ b1,
                       float* __restrict__ piT,
                       float* __restrict__ pjT) {
  const int lane = threadIdx.x;        // 0..31
  const int bi   = blockIdx.x;         // node tile   (0..127)
  const int bj   = blockIdx.y;         // hidden tile (0..7, over 128 concat)
  const int l15  = lane & 15;
  const int hi16 = lane >> 4;          // 0 or 1
  const int m    = bi * 16 + l15;      // node row this lane loads for A
  const int gn   = bj * 16 + l15;      // concat-hidden column (0..127) for B
  const int khalf = hi16 * 2;

  // Wcat[k][gn]: gn<64 -> W1[k][gn]; else W1[256+k][gn-64]
  const long woff = (gn < kH) ? (long)gn : ((long)(gn - kH) + (long)kD * kH);

  const float* xrow = X + (long)m * kD;
  v8f c = {};
  for (int k = 0; k < kD; k += 4) {
    const int k0 = k + khalf;
    v2f a = *(const v2f*)(xrow + k0);          // X[m][k0], X[m][k0+1]
    v2f b;
    b.x = W1[(long)k0 * kH + woff];            // Wcat[k0  ][gn]
    b.y = W1[(long)(k0 + 1) * kH + woff];      // Wcat[k0+1][gn]
    c = __builtin_amdgcn_wmma_f32_16x16x4_f32(
            /*neg_a=*/false, a, /*neg_b=*/false, b,
            /*c_mod=*/(short)0, c, /*reuse_a=*/false, /*reuse_b=*/false);
  }

  float badd;
  float* dst;
  if (gn < kH) { badd = b1[gn]; dst = piT + (long)gn * kN; }
  else         { badd = 0.0f;   dst = pjT + (long)(gn - kH) * kN; }
  const int rbase = bi * 16 + hi16 * 8;
  #pragma unroll
  for (int v = 0; v < 8; ++v) dst[rbase + v] = c[v] + badd;
}

// ---------------------------------------------------------------------------
// Kernel B: upper-triangular 64x64 tile pairs. Computes both orientations
// L[i,j] and L[j,i], symmetrizes, applies sigmoid, writes both outputs.
// LDS: 4 staged panels of [h][node] (64x64 f32 each) = 64 KB.
// ---------------------------------------------------------------------------
__global__ __launch_bounds__(256)
void edgedec_pair(const float* __restrict__ piT,
                  const float* __restrict__ pjT,
                  const float* __restrict__ W2,
                  const float* __restrict__ b2,
                  float* __restrict__ out) {
  const int bj = blockIdx.x;
  const int bi = blockIdx.y;
  if (bj < bi) return;                 // upper triangle only (halves h-work)

  __shared__ float sPiI[kH * 64];      // pi' rows of I-block, [h][r]
  __shared__ float sPjI[kH * 64];      // pj  rows of I-block
  __shared__ float sPiJ[kH * 64];      // pi' rows of J-block
  __shared__ float sPjJ[kH * 64];      // pj  rows of J-block

  const int tid   = threadIdx.x;
  const int baseI = bi * 64;
  const int baseJ = bj * 64;

  // 1024 float4 per panel; 256 threads x 4 chunks, fully coalesced.
  for (int f = tid; f < 1024; f += 256) {
    const int h  = f >> 4;
    const int cc = (f & 15) << 2;
    const long g = (long)h * kN;
    *(float4*)&sPiI[h * 64 + cc] = *(const float4*)&piT[g + baseI + cc];
    *(float4*)&sPjI[h * 64 + cc] = *(const float4*)&pjT[g + baseI + cc];
    *(float4*)&sPiJ[h * 64 + cc] = *(const float4*)&piT[g + baseJ + cc];
    *(float4*)&sPjJ[h * 64 + cc] = *(const float4*)&pjT[g + baseJ + cc];
  }
  __syncthreads();

  const int tx = tid & 15, ty = tid >> 4;
  const int i0 = ty * 4, j0 = tx * 4;

  float L1[4][4] = {};   // L[i,j] orientation (b1 already folded into pi')
  float L2[4][4] = {};   // L[j,i] orientation

  #pragma unroll 4
  for (int h = 0; h < kH; ++h) {
    const float w = W2[h];                               // uniform -> s_load
    float4 xpi = *(const float4*)&sPiI[h * 64 + i0];     // ds_load_b128
    float4 xpj = *(const float4*)&sPjI[h * 64 + i0];
    float4 ypi = *(const float4*)&sPiJ[h * 64 + j0];
    float4 ypj = *(const float4*)&sPjJ[h * 64 + j0];
    const float ai[4] = {xpi.x, xpi.y, xpi.z, xpi.w};    // pi'[i_u][h]
    const float aj[4] = {xpj.x, xpj.y, xpj.z, xpj.w};    // pj [i_u][h]
    const float ci[4] = {ypi.x, ypi.y, ypi.z, ypi.w};    // pi'[j_v][h]
    const float cj[4] = {ypj.x, ypj.y, ypj.z, ypj.w};    // pj [j_v][h]
    #pragma unroll
    for (int u = 0; u < 4; ++u) {
      #pragma unroll
      for (int v = 0; v < 4; ++v) {
        L1[u][v] += fmaxf(ai[u] + cj[v], 0.0f) * w;
        L2[u][v] += fmaxf(ci[v] + aj[u], 0.0f) * w;
      }
    }
  }

  const float bb = b2[0];
  #pragma unroll
  for (int u = 0; u < 4; ++u) {
    #pragma unroll
    for (int v = 0; v < 4; ++v) {
      const float lg = 0.5f * (L1[u][v] + L2[u][v]) + bb;
      const float s  = 1.0f / (1.0f + __expf(-lg));
      const int gi = baseI + i0 + u;
      const int gj = baseJ + j0 + v;
      out[(long)gi * kN + gj] = s;
      out[(long)gj * kN + gi] = s;   // mirror (same value by construction)
    }
  }
}

// ---------------------------------------------------------------------------
extern "C" void kernel_launch(void* const* d_in, const int* in_sizes, int n_in,
                              void* d_out, int out_size, void* d_ws, size_t ws_size,
                              hipStream_t stream) {
  const float* X  = (const float*)d_in[0];   // (2048, 256)
  const float* W1 = (const float*)d_in[1];   // (512, 64)
  const float* b1 = (const float*)d_in[2];   // (64,)
  const float* W2 = (const float*)d_in[3];   // (64, 1)
  const float* b2 = (const float*)d_in[4];   // (1,)
  float* out = (float*)d_out;                // (2048, 2048)

  float* piT = (float*)d_ws;                 // (64, 2048) transposed pi'
  float* pjT = piT + (long)kH * kN;          // (64, 2048) transposed pj

  dim3 gA(kN / 16, 128 / 16);                // (128, 8) 16x16 tiles
  edgedec_gemm_wmma<<<gA, 32, 0, stream>>>(X, W1, b1, piT, pjT);

  dim3 gB(kN / 64, kN / 64);                 // (32, 32), lower half early-exits
  edgedec_pair<<<gB, 256, 0, stream>>>(piT, pjT, W2, b2, out);
}